// SpatialTemporalConv_74431783240188
// MI455X (gfx1250) — compile-verified
//
#include <hip/hip_runtime.h>

// Problem constants (from reference)
#define BZd  32
#define CAPd 134
#define SEQd 144
#define INVd 16
#define ESP  1072
#define EFT  64
#define CAPP 144          // CAP padded to 9 x 16 tiles
#define COMW 48           // concat width (3 x INV)
#define FG   8            // graphs per block in k_feature

typedef float v2f __attribute__((ext_vector_type(2)));
typedef float v8f __attribute__((ext_vector_type(8)));

// D(16x16,f32) = A(16x4,f32) x B(4x16,f32) + C   -- full fp32 matrix op
__device__ __forceinline__ v8f wmma4(v2f a, v2f b, v8f c) {
  return __builtin_amdgcn_wmma_f32_16x16x4_f32(
      /*neg_a=*/false, a, /*neg_b=*/false, b,
      /*c_mod=*/(short)0, c, /*reuse_a=*/false, /*reuse_b=*/false);
}

// ---- CDNA5 async global->LDS staging (ASYNCcnt path), guarded -------------
#if defined(__gfx1250__) && __has_builtin(__builtin_amdgcn_global_load_async_to_lds_b128)
#define ASYNC_LDS 1
#endif

#ifdef ASYNC_LDS
// Builtin prototype (from compiler diagnostic): param0 = global int4*,
// param1 = shared int4*, then imm offset + imm cpol. Pointee must be the
// generic-vector (vector_size) flavor to match "V4i" exactly.
typedef int v4i_gv __attribute__((vector_size(16)));
typedef __attribute__((address_space(1))) v4i_gv* gbl_v4i_p;
typedef __attribute__((address_space(3))) v4i_gv* lds_v4i_p;
#endif

__device__ __forceinline__ void lds_stage16(void* ldsDst, const void* gSrc) {
#ifdef ASYNC_LDS
  __builtin_amdgcn_global_load_async_to_lds_b128(
      (gbl_v4i_p)(uintptr_t)gSrc,
      (lds_v4i_p)(uint32_t)(uintptr_t)ldsDst,
      0, 0);
#else
  *(float4*)ldsDst = *(const float4*)gSrc;
#endif
}

__device__ __forceinline__ void lds_stage_wait() {
#ifdef ASYNC_LDS
#if __has_builtin(__builtin_amdgcn_s_wait_asynccnt)
  __builtin_amdgcn_s_wait_asynccnt(0);
#else
  asm volatile("s_wait_asynccnt 0" ::: "memory");
#endif
#endif
}

// ---------------------------------------------------------------------------
// Kernel 1: spatial SAGE over CAP nodes. One block per (b,s). 9 waves.
// out[c][i] = mean[c][:] @ Wl + x[c][:] @ Wr + bl  -> com[row][0:16]
// ---------------------------------------------------------------------------
__global__ __launch_bounds__(288)
void k_spatial(const float* __restrict__ src, const int* __restrict__ ge,
               const float* __restrict__ Wl, const float* __restrict__ bl,
               const float* __restrict__ Wr, float* __restrict__ com) {
  __shared__ float xls[CAPP][INVd];
  __shared__ float agg[CAPP][INVd];     // becomes mean
  __shared__ float deg[CAPP];
  __shared__ float wls[2 * INVd][INVd]; // [Wl ; Wr] stacked (K=32)
  const int tid = threadIdx.x;
  const int b = blockIdx.x / SEQd;
  const int s = blockIdx.x % SEQd;

  for (int i = tid; i < CAPP * INVd; i += 288) {
    int c = i >> 4, f = i & 15;
    float v = 0.f;
    if (c < CAPd) v = src[((size_t)(b * CAPd + c) * SEQd + s) * INVd + f];
    xls[c][f] = v;
    agg[c][f] = 0.f;
  }
  for (int i = tid; i < CAPP; i += 288) deg[i] = 0.f;
  for (int i = tid; i < 2 * INVd * INVd; i += 288) {
    int k = i >> 4, n = i & 15;
    wls[k][n] = (k < INVd) ? Wl[k * INVd + n] : Wr[(k - INVd) * INVd + n];
  }
  __syncthreads();

  // scatter-add over edges (LDS float atomics -> ds_add_f32)
  for (int i = tid; i < ESP * INVd; i += 288) {
    int e = i >> 4, f = i & 15;
    int es = ge[e], ed = ge[ESP + e];
    atomicAdd(&agg[ed][f], xls[es][f]);
    if (f == 0) atomicAdd(&deg[ed], 1.0f);
  }
  __syncthreads();
  for (int i = tid; i < CAPP * INVd; i += 288) {
    int c = i >> 4, f = i & 15;
    agg[c][f] = agg[c][f] / fmaxf(deg[c], 1.0f);
  }
  __syncthreads();

  const int wave = tid >> 5, lane = tid & 31;
  const int m = lane & 15, half = lane >> 4;
  const int row = wave * 16 + m;            // A-fragment row (M = lane%16)
  v8f acc = {};
#pragma unroll
  for (int kk = 0; kk < 8; ++kk) {          // K = 32 (mean|x)
    int kb = kk * 4 + 2 * half;
    const float* ap = (kb < INVd) ? &agg[row][kb] : &xls[row][kb - INVd];
    v2f a;  a.x = ap[0];       a.y = ap[1];
    v2f bb; bb.x = wls[kb][m]; bb.y = wls[kb + 1][m];
    acc = wmma4(a, bb, acc);
  }
  float bias = bl[m];
#pragma unroll
  for (int r = 0; r < 8; ++r) {             // C: M = r + 8*half, N = m
    int c = wave * 16 + r + 8 * half;
    if (c < CAPd)
      com[((size_t)(b * CAPd + c) * SEQd + s) * COMW + m] = acc[r] + bias;
  }
}

// ---------------------------------------------------------------------------
// Kernel 2: feature SAGE, transposed: outT(144x16) = WT(144x288) @ XMT(288x16).
// A = weight fragments held in REGISTERS (per-wave M-tile of s_out), reused
// across FG graphs per block. Per-graph XM staged in LDS via async b128.
// out[i][s] -> com[row(n,s)][16 + i]
// ---------------------------------------------------------------------------
__global__ __launch_bounds__(288)
void k_feature(const float* __restrict__ src, const int* __restrict__ fe,
               const float* __restrict__ Wl, const float* __restrict__ bl,
               const float* __restrict__ Wr, float* __restrict__ com) {
  __shared__ float xls2[SEQd][INVd];   // x  in [seq][inv] (== global layout)
  __shared__ float mn2[SEQd][INVd];    // agg -> mean, same layout
  __shared__ float degr[INVd];         // 1 / max(deg, 1)
  const int tid = threadIdx.x;
  const int wave = tid >> 5, lane = tid & 31;
  const int i = lane & 15, half = lane >> 4;
  const int nt = wave * 16;            // s_out tile (M dimension)

  // A fragments (weights, transposed) -> registers, reused for all FG graphs
  v2f aW[72];
#pragma unroll
  for (int kk = 0; kk < 72; ++kk) {
    int kb = kk * 4 + 2 * half;
    if (kb < SEQd) {
      aW[kk].x = Wl[kb * SEQd + nt + i];
      aW[kk].y = Wl[(kb + 1) * SEQd + nt + i];
    } else {
      int k2 = kb - SEQd;
      aW[kk].x = Wr[k2 * SEQd + nt + i];
      aW[kk].y = Wr[(k2 + 1) * SEQd + nt + i];
    }
  }
  float bv[8];
#pragma unroll
  for (int r = 0; r < 8; ++r) bv[r] = bl[nt + r + 8 * half];

  // degree (graph structure is shared by all n)
  if (tid < INVd) degr[tid] = 0.f;
  __syncthreads();
  for (int e = tid; e < EFT; e += 288) atomicAdd(&degr[fe[EFT + e]], 1.0f);
  __syncthreads();
  if (tid < INVd) degr[tid] = 1.0f / fmaxf(degr[tid], 1.0f);
  __syncthreads();

  const int n0 = blockIdx.x * FG;
  for (int nn = 0; nn < FG; ++nn) {
    const int n = n0 + nn;
    // stage x (contiguous 9216 B) via async global->LDS b128
    const float* g = src + (size_t)n * SEQd * INVd;
    for (int idx = tid; idx < (SEQd * INVd) / 4; idx += 288)
      lds_stage16(&xls2[0][0] + idx * 4, g + idx * 4);
    for (int idx = tid; idx < SEQd * INVd; idx += 288) (&mn2[0][0])[idx] = 0.f;
    lds_stage_wait();
    __syncthreads();

    // scatter-add over the 64 feature edges (per seq position)
    for (int idx = tid; idx < EFT * SEQd; idx += 288) {
      int e = idx & (EFT - 1), sp = idx >> 6;
      atomicAdd(&mn2[sp][fe[EFT + e]], xls2[sp][fe[e]]);
    }
    __syncthreads();
    for (int idx = tid; idx < SEQd * INVd; idx += 288) {
      int sp = idx >> 4, f = idx & 15;
      mn2[sp][f] *= degr[f];
    }
    __syncthreads();

    v8f acc = {};
#pragma unroll
    for (int kk = 0; kk < 72; ++kk) {       // K = 288 ([mean;x])
      int kb = kk * 4 + 2 * half;
      const float* bp = (kb < SEQd) ? &mn2[kb][i] : &xls2[kb - SEQd][i];
      v2f b; b.x = bp[0]; b.y = bp[INVd];   // rows kb, kb+1
      acc = wmma4(aW[kk], b, acc);
    }
#pragma unroll
    for (int r = 0; r < 8; ++r) {           // C: M = s_out in tile, N = i
      int sOut = nt + r + 8 * half;
      com[((size_t)n * SEQd + sOut) * COMW + 16 + i] = acc[r] + bv[r];
    }
    __syncthreads();                        // protect LDS before restage
  }
}

// ---------------------------------------------------------------------------
// Kernel 3: temporal convs as 8 taps of 16x16 matmuls (K=128 total).
// One block per n; input staged via async b128 into [seq+halo][inv].
// out[o][h] -> com[row(n,h)][32 + o]
// ---------------------------------------------------------------------------
__global__ __launch_bounds__(288)
void k_temporal(const float* __restrict__ src, const float* __restrict__ w1,
                const float* __restrict__ w2, float* __restrict__ com) {
  __shared__ float X2[SEQd + 4][INVd];       // halo of 2 rows each side
  __shared__ float wt[8][INVd][INVd];        // [tap][o][i]
  const int tid = threadIdx.x;
  const int n = blockIdx.x;

  const float* g = src + (size_t)n * SEQd * INVd;
  for (int idx = tid; idx < (SEQd * INVd) / 4; idx += 288)
    lds_stage16(&X2[2][0] + idx * 4, g + idx * 4);
  if (tid < 2 * INVd) {                      // zero halo rows
    (&X2[0][0])[tid] = 0.f;
    (&X2[SEQd + 2][0])[tid] = 0.f;
  }
  for (int idx = tid; idx < 8 * INVd * INVd; idx += 288) {
    int t = idx >> 8, rem = idx & 255;
    int o = rem >> 4, ii = rem & 15;
    wt[t][o][ii] = (t < 3) ? w1[(o * INVd + ii) * 3 + t]
                           : w2[(o * INVd + ii) * 5 + (t - 3)];
  }
  lds_stage_wait();
  __syncthreads();

  const int wave = tid >> 5, lane = tid & 31;
  const int m = lane & 15, half = lane >> 4;
  const int ht = wave * 16;
  const int offs[8] = {-1, 0, 1, -2, -1, 0, 1, 2}; // conv1 k-1, conv2 k-2
  v8f acc = {};
#pragma unroll
  for (int t = 0; t < 8; ++t) {
    int hsrc = ht + m + offs[t] + 2;
#pragma unroll
    for (int kk = 0; kk < 4; ++kk) {          // K = 16 channels per tap
      int kb = kk * 4 + 2 * half;
      v2f a, b;
      a.x = wt[t][m][kb];  a.y = wt[t][m][kb + 1];
      const float* bp = &X2[hsrc][kb];        // contiguous pair
      b.x = bp[0]; b.y = bp[1];
      acc = wmma4(a, b, acc);
    }
  }
#pragma unroll
  for (int r = 0; r < 8; ++r) {
    int o = r + 8 * half;
    int h = ht + m;
    com[((size_t)n * SEQd + h) * COMW + 32 + o] = acc[r];
  }
}

// ---------------------------------------------------------------------------
// Kernel 4: out = src + com(617472x48) @ fc_w(48x16) + fc_b. 8 waves/block.
// ---------------------------------------------------------------------------
__global__ __launch_bounds__(256)
void k_fc(const float* __restrict__ src, const float* __restrict__ com,
          const float* __restrict__ fcw, const float* __restrict__ fcb,
          float* __restrict__ out) {
  __shared__ float wls[COMW][INVd];
  const int tid = threadIdx.x;
  for (int i = tid; i < COMW * INVd; i += 256) (&wls[0][0])[i] = fcw[i];
  __syncthreads();

  const int wave = tid >> 5, lane = tid & 31;
  const int m = lane & 15, half = lane >> 4;
  const size_t rowbase = ((size_t)blockIdx.x * 8 + wave) * 16;
  v8f acc = {};
#pragma unroll
  for (int kk = 0; kk < 12; ++kk) {          // K = 48
    int kb = kk * 4 + 2 * half;
    const float* ap = &com[(rowbase + m) * COMW + kb]; // contiguous pair
    v2f a;  a.x = ap[0];       a.y = ap[1];
    v2f bb; bb.x = wls[kb][m]; bb.y = wls[kb + 1][m];
    acc = wmma4(a, bb, acc);
  }
  float bias = fcb[m];
#pragma unroll
  for (int r = 0; r < 8; ++r) {
    size_t row = rowbase + r + 8 * half;
    size_t idx = row * INVd + m;
    out[idx] = src[idx] + acc[r] + bias;
  }
}

// ---------------------------------------------------------------------------
extern "C" void kernel_launch(void* const* d_in, const int* in_sizes, int n_in,
                              void* d_out, int out_size, void* d_ws, size_t ws_size,
                              hipStream_t stream) {
  const float* src = (const float*)d_in[0];
  const int*   ge  = (const int*)d_in[1];
  const int*   fe  = (const int*)d_in[2];
  const float* gWl = (const float*)d_in[3];
  const float* gbl = (const float*)d_in[4];
  const float* gWr = (const float*)d_in[5];
  const float* fWl = (const float*)d_in[6];
  const float* fbl = (const float*)d_in[7];
  const float* fWr = (const float*)d_in[8];
  const float* w1  = (const float*)d_in[9];
  const float* w2  = (const float*)d_in[10];
  const float* fcw = (const float*)d_in[11];
  const float* fcb = (const float*)d_in[12];
  float* out = (float*)d_out;
  float* com = (float*)d_ws;   // (BZ*CAP*SEQ) x 48 concat scratch (~119 MB)

  k_spatial <<<BZd * SEQd, 288, 0, stream>>>(src, ge, gWl, gbl, gWr, com);
  k_feature <<<(BZd * CAPd) / FG, 288, 0, stream>>>(src, fe, fWl, fbl, fWr, com);
  k_temporal<<<BZd * CAPd, 288, 0, stream>>>(src, w1, w2, com);
  const int rowTiles = (BZd * CAPd * SEQd) / 16;   // 38592
  k_fc      <<<rowTiles / 8, 256, 0, stream>>>(src, com, fcw, fcb, out);
}